// Hash4Encoder_37830071943725
// MI455X (gfx1250) — compile-verified
//
#include <hip/hip_runtime.h>
#include <hip/hip_bf16.h>

#define TPB      256
#define NLEVELS  16
#define PRIME1   2654435761u
#define PRIME2   805459861u
#define PRIME3   3674653429u

// LDS row stride (pad 32 -> 33 words): conflict-free for both the per-thread
// row writes (stride 33, gcd(33,64)=1) and the transposed float4 read phase.
#define ROWPAD   33

typedef float fvec4 __attribute__((ext_vector_type(4)));
typedef float fvec2 __attribute__((ext_vector_type(2)));

__global__ __launch_bounds__(TPB)
void hash4_encode_kernel(const float* __restrict__ xyzts,       // [B,4]
                         const float* __restrict__ table,
                         const int*   __restrict__ shapes,      // [16][4]
                         const int*   __restrict__ sizes,       // [16]
                         const int*   __restrict__ indicator,   // [16]
                         const int*   __restrict__ offsets,     // [16]
                         float*       __restrict__ out,         // [B, 32]
                         int npoints)
{
    __shared__ float lds[TPB * ROWPAD];

    const int t  = threadIdx.x;
    const int b0 = blockIdx.x * TPB;
    const int b  = b0 + t;

    // Clamp so tail-block threads do safe (duplicate) work; stores are
    // bounds-guarded in the writeout phase.
    const int   bl = (b < npoints) ? b : (npoints > 0 ? npoints - 1 : 0);
    const fvec4 p  = __builtin_nontemporal_load(
                         reinterpret_cast<const fvec4*>(xyzts) + bl);  // streamed once

    float* myrow = &lds[t * ROWPAD];

    #pragma unroll 1
    for (int l = 0; l < NLEVELS; ++l) {
        // Uniform per-level config -> scalar loads / SGPRs.
        const int4     sh   = *reinterpret_cast<const int4*>(shapes + 4 * l);
        const unsigned size = (unsigned)sizes[l];
        const int      ind  = indicator[l];
        const int      off  = offsets[l];

        const float rx = (float)sh.x, ry = (float)sh.y, rz = (float)sh.z, rt = (float)sh.w;

        const float px = p.x * rx, py = p.y * ry, pz = p.z * rz, pt = p.w * rt;
        const float gx = fminf(fmaxf(floorf(px), 0.f), rx - 1.f);
        const float gy = fminf(fmaxf(floorf(py), 0.f), ry - 1.f);
        const float gz = fminf(fmaxf(floorf(pz), 0.f), rz - 1.f);
        const float gt = fminf(fmaxf(floorf(pt), 0.f), rt - 1.f);
        const float fxp = fminf(fmaxf(px - gx, 0.f), 1.f);
        const float fyp = fminf(fmaxf(py - gy, 0.f), 1.f);
        const float fzp = fminf(fmaxf(pz - gz, 0.f), 1.f);
        const float ftp = fminf(fmaxf(pt - gt, 0.f), 1.f);
        const unsigned ux = (unsigned)gx, uy = (unsigned)gy;
        const unsigned uz = (unsigned)gz, ut = (unsigned)gt;

        // Pairwise quadrilinear weights: w[c] = wxy[c&3] * wzt[c>>2]
        float wxy[4], wzt[4];
        {
            const float x0 = 1.f - fxp, y0 = 1.f - fyp, z0 = 1.f - fzp, t0 = 1.f - ftp;
            wxy[0] = x0 * y0;  wxy[1] = fxp * y0;  wxy[2] = x0 * fyp;  wxy[3] = fxp * fyp;
            wzt[0] = z0 * t0;  wzt[1] = fzp * t0;  wzt[2] = z0 * ftp;  wzt[3] = fzp * ftp;
        }

        float s0a = 0.f, s1a = 0.f, s0b = 0.f, s1b = 0.f;

        if (ind == 1) {
            // Dense under-hash: row-major index, no modulo needed (index < size
            // by construction: clamped grid + 1 stays inside res+1 box).
            const unsigned st1 = (unsigned)(sh.x + 1);
            const unsigned st2 = st1 * (unsigned)(sh.y + 1);
            const unsigned st3 = st2 * (unsigned)(sh.z + 1);
            const unsigned ax0 = ux,       ax1 = ux + 1u;
            const unsigned ay0 = uy * st1, ay1 = ay0 + st1;
            const unsigned az0 = uz * st2, az1 = az0 + st2;
            const unsigned at0 = ut * st3, at1 = at0 + st3;
            unsigned c01[4], c23[4];
            c01[0] = ax0 + ay0; c01[1] = ax1 + ay0; c01[2] = ax0 + ay1; c01[3] = ax1 + ay1;
            c23[0] = az0 + at0; c23[1] = az1 + at0; c23[2] = az0 + at1; c23[3] = az1 + at1;
            #pragma unroll
            for (int c = 0; c < 16; ++c) {
                const unsigned h = c01[c & 3] + c23[c >> 2];
                const fvec2 f = *reinterpret_cast<const fvec2*>(table + (off + (int)(2u * h)));
                const float w = wxy[c & 3] * wzt[c >> 2];
                if (c & 1) { s0b = fmaf(w, f.x, s0b); s1b = fmaf(w, f.y, s1b); }
                else       { s0a = fmaf(w, f.x, s0a); s1a = fmaf(w, f.y, s1a); }
            }
        } else {
            // fast_hash: xor of coord*prime (uint32 wraparound), then % size.
            const unsigned ax0 = ux,          ax1 = ux + 1u;
            const unsigned ay0 = uy * PRIME1, ay1 = ay0 + PRIME1;
            const unsigned az0 = uz * PRIME2, az1 = az0 + PRIME2;
            const unsigned at0 = ut * PRIME3, at1 = at0 + PRIME3;
            unsigned c01[4], c23[4];
            c01[0] = ax0 ^ ay0; c01[1] = ax1 ^ ay0; c01[2] = ax0 ^ ay1; c01[3] = ax1 ^ ay1;
            c23[0] = az0 ^ at0; c23[1] = az1 ^ at0; c23[2] = az0 ^ at1; c23[3] = az1 ^ at1;
            const unsigned mask = size - 1u;
            if ((size & (size - 1u)) == 0u) {   // uniform branch; MAX_MAP = 2^19
                #pragma unroll
                for (int c = 0; c < 16; ++c) {
                    const unsigned h = (c01[c & 3] ^ c23[c >> 2]) & mask;
                    const fvec2 f = *reinterpret_cast<const fvec2*>(table + (off + (int)(2u * h)));
                    const float w = wxy[c & 3] * wzt[c >> 2];
                    if (c & 1) { s0b = fmaf(w, f.x, s0b); s1b = fmaf(w, f.y, s1b); }
                    else       { s0a = fmaf(w, f.x, s0a); s1a = fmaf(w, f.y, s1a); }
                }
            } else {
                #pragma unroll
                for (int c = 0; c < 16; ++c) {
                    const unsigned h = (c01[c & 3] ^ c23[c >> 2]) % size;
                    const fvec2 f = *reinterpret_cast<const fvec2*>(table + (off + (int)(2u * h)));
                    const float w = wxy[c & 3] * wzt[c >> 2];
                    if (c & 1) { s0b = fmaf(w, f.x, s0b); s1b = fmaf(w, f.y, s1b); }
                    else       { s0a = fmaf(w, f.x, s0a); s1a = fmaf(w, f.y, s1a); }
                }
            }
        }

        myrow[2 * l]     = s0a + s0b;
        myrow[2 * l + 1] = s1a + s1b;
    }

    __syncthreads();

    // Coalesced transposed writeout: block's output region is contiguous
    // [b0*32, (b0+TPB)*32). Each thread streams 8 float4's (non-temporal).
    const long long total = (long long)npoints * 32;
    const long long base  = (long long)b0 * 32;
    #pragma unroll
    for (int k = 0; k < 8; ++k) {
        const int       j  = 4 * t + 1024 * k;       // 0..8191
        const long long gj = base + j;
        const int r = j >> 5;
        const int c = j & 31;
        if (gj + 3 < total) {
            fvec4 v;
            v.x = lds[r * ROWPAD + c + 0];
            v.y = lds[r * ROWPAD + c + 1];
            v.z = lds[r * ROWPAD + c + 2];
            v.w = lds[r * ROWPAD + c + 3];
            __builtin_nontemporal_store(v, reinterpret_cast<fvec4*>(out + gj));
        } else if (gj < total) {
            #pragma unroll 1
            for (int m = 0; m < 4; ++m)
                if (gj + m < total) out[gj + m] = lds[r * ROWPAD + c + m];
        }
    }
}

extern "C" void kernel_launch(void* const* d_in, const int* in_sizes, int n_in,
                              void* d_out, int out_size, void* d_ws, size_t ws_size,
                              hipStream_t stream) {
    (void)n_in; (void)out_size; (void)d_ws; (void)ws_size;
    const float* xyzts     = (const float*)d_in[0];
    const float* table     = (const float*)d_in[1];
    const int*   shapes    = (const int*)d_in[2];
    const int*   sizes     = (const int*)d_in[3];
    const int*   indicator = (const int*)d_in[4];
    const int*   offsets   = (const int*)d_in[5];
    float*       out       = (float*)d_out;

    const int npoints = in_sizes[0] / 4;
    const int blocks  = (npoints + TPB - 1) / TPB;
    if (blocks > 0) {
        hipLaunchKernelGGL(hash4_encode_kernel, dim3(blocks), dim3(TPB), 0, stream,
                           xyzts, table, shapes, sizes, indicator, offsets, out, npoints);
    }
}